// HarmonicOscillator_30889404793031
// MI455X (gfx1250) — compile-verified
//
#include <hip/hip_runtime.h>
#include <math.h>

// HarmonicOscillator on gfx1250 (MI455X):
//  - phase offsets per 256-sample segment precomputed (segment sums + double-precision serial scan)
//  - main kernel: TDM (tensor_load_to_lds) stages the 3x101 harmonic-distribution tile into LDS,
//    Chebyshev sin recurrence generates the Sin[16x4] A-tiles, V_WMMA_F32_16X16X4_F32 computes
//    D[16 samples x 3 frames] = Sin x nhd^T, final lerp+amp written to global.
//
// WMMA layout assumptions (per CDNA5 ISA 7.12.2):
//   A (16x4 f32): lanes 0-15 row M=lane, VGPR0=K0/VGPR1=K1; lanes 16-31 same rows, K2/K3.  [ISA table]
//   B (4x16 f32): assumed mirror: lanes 0-15 N=lane with K0/K1, lanes 16-31 N=lane-16 with K2/K3.
//   C/D (16x16 f32): VGPR r: lanes 0-15 -> (M=r, N=lane); lanes 16-31 -> (M=r+8, N=lane-16). [ISA table]

#define NB   8      // batch
#define NF   400    // frames
#define NH   101    // harmonics
#define HOP  256
#define NT   (NF*HOP)
#define DF_SCALE 3.9269908169872414e-4f   // 2*pi/16000

typedef float v2f  __attribute__((ext_vector_type(2)));
typedef float v8f  __attribute__((ext_vector_type(8)));
typedef unsigned int u32x4 __attribute__((ext_vector_type(4)));
typedef int   i32x4 __attribute__((ext_vector_type(4)));
typedef int   i32x8 __attribute__((ext_vector_type(8)));

// torch/jax linear upsample (align_corners=False) source position for sample j of segment k
__device__ __forceinline__ void frame_pos(int k, int j, int& lo, int& hi, float& w) {
    float src = (float)k + ((float)j + 0.5f) * (1.0f / 256.0f) - 0.5f;
    src = fminf(fmaxf(src, 0.0f), (float)(NF - 1));
    lo = (int)floorf(src);
    if (lo > NF - 1) lo = NF - 1;
    hi = lo + 1; if (hi > NF - 1) hi = NF - 1;
    w = src - (float)lo;
}

// ---------------- phase 1: per-segment sum of 2*pi*f/SR ----------------
__global__ void __launch_bounds__(256) segsum_kernel(const float* __restrict__ f0,
                                                     float* __restrict__ segsum) {
    __shared__ float red[256];
    const int k = blockIdx.x, b = blockIdx.y, j = threadIdx.x;
    int lo, hi; float w;
    frame_pos(k, j, lo, hi, w);
    const float* f0b = f0 + (size_t)b * NF;
    float f = (1.0f - w) * f0b[lo] + w * f0b[hi];
    red[j] = f * DF_SCALE;
    __syncthreads();
    for (int s = 128; s > 0; s >>= 1) {
        if (j < s) red[j] += red[j + s];
        __syncthreads();
    }
    if (j == 0) segsum[(size_t)b * NF + k] = red[0];
}

// ---------------- phase 2: exclusive scan of segment sums (double accum) ----------------
__global__ void offsets_kernel(const float* __restrict__ segsum, float* __restrict__ offs) {
    int b = threadIdx.x;
    if (b < NB) {
        double run = 0.0;
        for (int k = 0; k < NF; ++k) {
            offs[(size_t)b * NF + k] = (float)run;
            run += (double)segsum[(size_t)b * NF + k];
        }
    }
}

// ---------------- phase 3: synthesis ----------------
__global__ void __launch_bounds__(256) synth_kernel(const float* __restrict__ nhd,
                                                    const float* __restrict__ amp,
                                                    const float* __restrict__ f0,
                                                    const float* __restrict__ offs,
                                                    float* __restrict__ out) {
    __shared__ float s_tmp[3 * NH + 1];   // raw TDM landing buffer (3 rows x 101)
    __shared__ float s_nhd[3 * 104];      // masked, zero-padded to K=104
    __shared__ float s_f0[3];
    __shared__ float s_amp[3];
    __shared__ float s_omega[256];
    __shared__ float s_D[256 * 3];        // D[sample][frame-col]

    const int k   = blockIdx.x;
    const int b   = blockIdx.y;
    const int tid = threadIdx.x;
    const int r0  = min(max(k - 1, 0), NF - 3);   // first of 3 frames staged

    if (tid < 3) {
        s_f0[tid]  = f0[(size_t)b * NF + r0 + tid];
        s_amp[tid] = amp[(size_t)b * NF + r0 + tid];
    }

    // ---- TDM: DMA the 3x101 f32 tile of n_harm_dis into LDS (wave 0 issues) ----
    if (tid < 32) {
        unsigned long long ga =
            (unsigned long long)(const void*)(nhd + ((size_t)b * NF + r0) * NH);
        unsigned int ldsa = (unsigned int)(unsigned long long)(const void*)s_tmp;
        // D# group 0: count=1, lds_addr, global_addr[56:0], type=2
        u32x4 g0 = { 1u,
                     ldsa,
                     (unsigned int)(ga & 0xffffffffu),
                     (unsigned int)((ga >> 32) & 0x1ffffffu) | (2u << 30) };
        // D# group 1: data_size=4B(code 2); tensor 101x3; tile 101x3; dim0 stride 101
        i32x8 g1 = { (int)(2u << 16),      // workgroup_mask=0, data_size=2
                     (int)(101u << 16),    // tensor_dim0[15:0]
                     (int)(3u   << 16),    // tensor_dim0[31:16]=0 | tensor_dim1[15:0]=3
                     (int)(101u << 16),    // tensor_dim1[31:16]=0 | tile_dim0=101
                     3,                    // tile_dim1=3, tile_dim2=0
                     101,                  // tensor_dim0_stride[31:0]
                     0, 0 };
        i32x4 gz4 = { 0, 0, 0, 0 };       // groups 2/3 unused (2D tensor)
        i32x8 gz8 = { 0, 0, 0, 0, 0, 0, 0, 0 };
        __builtin_amdgcn_tensor_load_to_lds(g0, g1, gz4, gz4, gz8, 0);
        __builtin_amdgcn_s_wait_tensorcnt(0);
    }
    __syncthreads();

    // ---- Nyquist mask (frame-rate, pre-upsample) + zero-pad rows to 104 ----
    for (int idx = tid; idx < 3 * 104; idx += 256) {
        int r = idx / 104, c = idx - r * 104;
        float v = 0.0f;
        if (c < NH) {
            float pitch = s_f0[r] * (float)(c + 1);
            v = (pitch >= 8000.0f) ? 0.0f : s_tmp[r * NH + c];
        }
        s_nhd[idx] = v;
    }

    // ---- per-sample interpolation weights + local inclusive scan for omega ----
    int lo, hi; float w;
    frame_pos(k, tid, lo, hi, w);
    const int li = lo - r0, hiI = hi - r0;
    float fj   = (1.0f - w) * s_f0[li]  + w * s_f0[hiI];
    float ampv = (1.0f - w) * s_amp[li] + w * s_amp[hiI];

    float v = fj * DF_SCALE;
    s_omega[tid] = v;
    __syncthreads();
    for (int off = 1; off < 256; off <<= 1) {
        float t = (tid >= off) ? s_omega[tid - off] : 0.0f;
        __syncthreads();
        v += t;
        s_omega[tid] = v;
        __syncthreads();
    }
    s_omega[tid] = offs[(size_t)b * NF + k] + v;
    __syncthreads();

    // ---- WMMA: D[16 samples x 16 cols] += Sin[16x4] x nhd^T[4x16], K = 0..103 ----
    const int lane = tid & 31;
    const int wv   = tid >> 5;
    const int n    = lane & 15;                 // B/D column (frames 0..2 useful)
    const int kb   = (lane < 16) ? 0 : 2;       // K sub-offset this half-wave supplies
    const float* brow = s_nhd + ((n < 3) ? n : 0) * 104;
    const float  bsc  = (n < 3) ? 1.0f : 0.0f;

    for (int g = 0; g < 2; ++g) {
        const int srow = wv * 32 + g * 16 + (lane & 15);
        const float x = s_omega[srow];
        // Chebyshev recurrence along K (stride 4): s(i+1) = 2cos(4x)*s(i) - s(i-1)
        const float mul2 = 2.0f * cosf(4.0f * x);
        float sa_c = sinf(x * (float)(kb + 1));   // h = 4i + kb + 1
        float sa_p = sinf(x * (float)(kb - 3));
        float sb_c = sinf(x * (float)(kb + 2));   // h = 4i + kb + 2
        float sb_p = sinf(x * (float)(kb - 2));

        v8f acc = { 0.f, 0.f, 0.f, 0.f, 0.f, 0.f, 0.f, 0.f };
        for (int i = 0; i < 26; ++i) {
            const int K0 = 4 * i + kb;
            v2f A;  A.x  = sa_c;            A.y  = sb_c;
            v2f Bv; Bv.x = bsc * brow[K0];  Bv.y = bsc * brow[K0 + 1];
            acc = __builtin_amdgcn_wmma_f32_16x16x4_f32(
                false, A, false, Bv, (short)0, acc, false, false);
            float ta = mul2 * sa_c - sa_p; sa_p = sa_c; sa_c = ta;
            float tb = mul2 * sb_c - sb_p; sb_p = sb_c; sb_c = tb;
        }
        if (n < 3) {
            const int mbase = wv * 32 + g * 16 + ((lane < 16) ? 0 : 8);
#pragma unroll
            for (int r = 0; r < 8; ++r)
                s_D[(mbase + r) * 3 + n] = acc[r];
        }
    }
    __syncthreads();

    // ---- combine: signal = amp * ((1-w)*D[:,lo] + w*D[:,hi]) ----
    float d0 = s_D[tid * 3 + li];
    float d1 = s_D[tid * 3 + hiI];
    out[(size_t)b * NT + (size_t)k * HOP + tid] = ampv * ((1.0f - w) * d0 + w * d1);
}

extern "C" void kernel_launch(void* const* d_in, const int* in_sizes, int n_in,
                              void* d_out, int out_size, void* d_ws, size_t ws_size,
                              hipStream_t stream) {
    const float* nhd = (const float*)d_in[0];   // [8,400,101]
    const float* amp = (const float*)d_in[1];   // [8,400,1]
    const float* f0  = (const float*)d_in[2];   // [8,400,1]
    float* out = (float*)d_out;                 // [8,102400,1]

    float* segsum = (float*)d_ws;               // 8*400 floats
    float* offs   = segsum + NB * NF;           // 8*400 floats

    segsum_kernel<<<dim3(NF, NB), 256, 0, stream>>>(f0, segsum);
    offsets_kernel<<<1, 32, 0, stream>>>(segsum, offs);
    synth_kernel<<<dim3(NF, NB), 256, 0, stream>>>(nhd, amp, f0, offs, out);
}